// TransformerShell_35631048688008
// MI455X (gfx1250) — compile-verified
//
#include <hip/hip_runtime.h>
#include <math.h>

// Problem constants (match reference)
#define NN 20000
#define EE 320000
#define CC 128
#define HH 4
#define HC (HH*CC)       // 512
#define EPSV 1e-5f

typedef __attribute__((ext_vector_type(2))) float v2f;
typedef __attribute__((ext_vector_type(8))) float v8f;

// ---- float <-> order-preserving int (for atomic max on f32) ----
__device__ __forceinline__ int f2ord(float f) {
  int i = __float_as_int(f);
  return (i >= 0) ? i : (i ^ 0x7FFFFFFF);
}
__device__ __forceinline__ float ord2f(int i) {
  return __int_as_float((i >= 0) ? i : (i ^ 0x7FFFFFFF));
}

// =====================================================================
// 0) init: zero accumulators, set running-max to -inf-ordinal
// =====================================================================
__global__ void ts_init_kernel(float* __restrict__ asum, int* __restrict__ amaxOrd,
                               float* __restrict__ accum) {
  size_t stride = (size_t)gridDim.x * blockDim.x;
  size_t total = (size_t)NN * HC;
  for (size_t t = (size_t)blockIdx.x * blockDim.x + threadIdx.x; t < total; t += stride) {
    accum[t] = 0.0f;
    if (t < (size_t)NN * HH) { asum[t] = 0.0f; amaxOrd[t] = (int)0x80000000; }
  }
}

// =====================================================================
// 1) WMMA fp32 GEMM:  Y[Nrows, M] = X[Nrows, 128] @ W[M,128]^T + bias[M]
//    Uses V_WMMA_F32_16X16X4_F32 (exact fp32 path on CDNA5).
//    Block: 128 threads = 4 waves; each block = 16 rows x 64 cols.
//    grid.x = Nrows/16, grid.y = M/64.
// =====================================================================
__global__ __launch_bounds__(128)
void ts_wmma_gemm(const float* __restrict__ X, const float* __restrict__ W,
                  const float* __restrict__ bias, float* __restrict__ Y, int M) {
  __shared__ float xs[16 * 132];          // 16x128 tile, stride 132 -> conflict-free
  const int tid = threadIdx.x;
  const int rowBase = blockIdx.x * 16;

  // Stage X tile (16x128 = 512 float4) into LDS
  #pragma unroll
  for (int it = 0; it < 4; ++it) {
    int f = (it * 128 + tid) * 4;         // 0..2044 step 4
    int r = f >> 7;
    int c = f & 127;
    float4 xv = *reinterpret_cast<const float4*>(X + (size_t)(rowBase + r) * CC + c);
    *reinterpret_cast<float4*>(&xs[r * 132 + c]) = xv;
  }
  __syncthreads();

  const int lane = tid & 31;
  const int wave = tid >> 5;
  const int hi   = lane >> 4;             // half-wave: selects K pair
  const int l15  = lane & 15;
  const int col  = blockIdx.y * 64 + wave * 16 + l15;   // output column (N dim of tile)
  const float* wrow = W + (size_t)col * CC;

  v8f acc = {};
  #pragma unroll
  for (int k = 0; k < CC; k += 4) {
    const int kk = k + 2 * hi;
    // A fragment (16x4): lane l15 = row, VGPRs = K={2*hi, 2*hi+1}
    float2 av = *reinterpret_cast<const float2*>(&xs[l15 * 132 + kk]);
    // B fragment (4x16): lane l15 = col, VGPRs = K={2*hi, 2*hi+1}; B[k][n] = W[n][k]
    float2 wv = *reinterpret_cast<const float2*>(wrow + kk);
    v2f a; a.x = av.x; a.y = av.y;
    v2f b; b.x = wv.x; b.y = wv.y;
    acc = __builtin_amdgcn_wmma_f32_16x16x4_f32(false, a, false, b, (short)0, acc,
                                                false, false);
  }

  const float bv = bias[col];
  const int mrow = hi * 8;                // D: VGPR j -> row j (lo half) / 8+j (hi half)
  #pragma unroll
  for (int j = 0; j < 8; ++j) {
    Y[(size_t)(rowBase + mrow + j) * M + col] = acc[j] + bv;
  }
}

// =====================================================================
// 2) edge pass A: alpha[e,h] = q[dst]·(k[src] + ea*We) / sqrt(C); running max
//    One wave per edge; 32 lanes x float4 cover C=128.
// =====================================================================
__global__ __launch_bounds__(256)
void ts_edge_alpha(const int* __restrict__ ei, const float* __restrict__ eattr,
                   const float* __restrict__ q, const float* __restrict__ k,
                   const float* __restrict__ We,
                   float* __restrict__ alpha, int* __restrict__ amaxOrd) {
  const int e = (int)(((size_t)blockIdx.x * blockDim.x + threadIdx.x) >> 5);
  const int lane = threadIdx.x & 31;
  if (e >= EE) return;
  const int src = ei[e];
  const int dst = ei[EE + e];
  const float ea = eattr[e];
  const float scale = 0.08838834764831845f;  // 1/sqrt(128)

  #pragma unroll
  for (int h = 0; h < HH; ++h) {
    const size_t co = (size_t)h * CC + lane * 4;
    float4 qv = *reinterpret_cast<const float4*>(q + (size_t)dst * HC + co);
    float4 kv = *reinterpret_cast<const float4*>(k + (size_t)src * HC + co);
    float4 wv = *reinterpret_cast<const float4*>(We + co);
    float p = qv.x * (kv.x + ea * wv.x) + qv.y * (kv.y + ea * wv.y)
            + qv.z * (kv.z + ea * wv.z) + qv.w * (kv.w + ea * wv.w);
    #pragma unroll
    for (int off = 16; off > 0; off >>= 1) p += __shfl_xor(p, off, 32);
    if (lane == 0) {
      const float a = p * scale;
      alpha[(size_t)e * HH + h] = a;
      atomicMax(&amaxOrd[(size_t)dst * HH + h], f2ord(a));
    }
  }
}

// =====================================================================
// 3) edge pass B: aexp = exp(alpha - amax[dst]); accumulate
//    asum[dst,h] += aexp;  accum[dst,h,:] += aexp*(v[src,h,:] + ea*We[h,:])
//    (normalization deferred to the node pass -> only two edge sweeps)
// =====================================================================
__global__ __launch_bounds__(256)
void ts_edge_scatter(const int* __restrict__ ei, const float* __restrict__ eattr,
                     const float* __restrict__ v, const float* __restrict__ We,
                     const float* __restrict__ alpha, const int* __restrict__ amaxOrd,
                     float* __restrict__ asum, float* __restrict__ accum) {
  const int e = (int)(((size_t)blockIdx.x * blockDim.x + threadIdx.x) >> 5);
  const int lane = threadIdx.x & 31;
  if (e >= EE) return;
  const int src = ei[e];
  const int dst = ei[EE + e];
  const float ea = eattr[e];

  #pragma unroll
  for (int h = 0; h < HH; ++h) {
    const float a  = alpha[(size_t)e * HH + h];
    const float mx = ord2f(amaxOrd[(size_t)dst * HH + h]);
    const float aexp = __expf(a - mx);
    if (lane == 0) atomicAdd(&asum[(size_t)dst * HH + h], aexp);
    const size_t co = (size_t)h * CC + lane * 4;
    float4 vv = *reinterpret_cast<const float4*>(v + (size_t)src * HC + co);
    float4 wv = *reinterpret_cast<const float4*>(We + co);
    float* o = accum + (size_t)dst * HC + co;
    atomicAdd(o + 0, aexp * (vv.x + ea * wv.x));
    atomicAdd(o + 1, aexp * (vv.y + ea * wv.y));
    atomicAdd(o + 2, aexp * (vv.z + ea * wv.z));
    atomicAdd(o + 3, aexp * (vv.w + ea * wv.w));
  }
}

// =====================================================================
// 4) node pass: pre = mean_h(accum/asum) + skip ; h0 = x + LN(pre; g0,b0)
//    One wave per node; lane holds 4 channels.
// =====================================================================
__global__ __launch_bounds__(128)
void ts_node_combine(const float* __restrict__ x, const float* __restrict__ skip,
                     const float* __restrict__ accum, const float* __restrict__ asum,
                     const float* __restrict__ g0, const float* __restrict__ b0,
                     float* __restrict__ h0) {
  const int n = (int)(((size_t)blockIdx.x * blockDim.x + threadIdx.x) >> 5);
  const int lane = threadIdx.x & 31;
  if (n >= NN) return;

  float inv[HH];
  #pragma unroll
  for (int h = 0; h < HH; ++h) {
    const float s = asum[(size_t)n * HH + h];
    inv[h] = (s > 0.0f) ? (0.25f / s) : 0.0f;   // fold head-mean 1/H into inv
  }

  float4 sk = *reinterpret_cast<const float4*>(skip + (size_t)n * CC + lane * 4);
  float pre0 = sk.x, pre1 = sk.y, pre2 = sk.z, pre3 = sk.w;
  #pragma unroll
  for (int h = 0; h < HH; ++h) {
    float4 av = *reinterpret_cast<const float4*>(accum + (size_t)n * HC + h * CC + lane * 4);
    pre0 += inv[h] * av.x; pre1 += inv[h] * av.y;
    pre2 += inv[h] * av.z; pre3 += inv[h] * av.w;
  }

  float s  = pre0 + pre1 + pre2 + pre3;
  float s2 = pre0 * pre0 + pre1 * pre1 + pre2 * pre2 + pre3 * pre3;
  #pragma unroll
  for (int off = 16; off > 0; off >>= 1) {
    s  += __shfl_xor(s,  off, 32);
    s2 += __shfl_xor(s2, off, 32);
  }
  const float mu   = s * (1.0f / 128.0f);
  const float var  = s2 * (1.0f / 128.0f) - mu * mu;
  const float rstd = rsqrtf(var + EPSV);

  float4 xv = *reinterpret_cast<const float4*>(x  + (size_t)n * CC + lane * 4);
  float4 gv = *reinterpret_cast<const float4*>(g0 + lane * 4);
  float4 bv = *reinterpret_cast<const float4*>(b0 + lane * 4);
  float4 r;
  r.x = xv.x + (pre0 - mu) * rstd * gv.x + bv.x;
  r.y = xv.y + (pre1 - mu) * rstd * gv.y + bv.y;
  r.z = xv.z + (pre2 - mu) * rstd * gv.z + bv.z;
  r.w = xv.w + (pre3 - mu) * rstd * gv.w + bv.w;
  *reinterpret_cast<float4*>(h0 + (size_t)n * CC + lane * 4) = r;
}

// =====================================================================
// 5) final: out = h0 + LN(lin; g1, b1)
// =====================================================================
__global__ __launch_bounds__(128)
void ts_final_ln(const float* __restrict__ h0, const float* __restrict__ lin,
                 const float* __restrict__ g1, const float* __restrict__ b1,
                 float* __restrict__ out) {
  const int n = (int)(((size_t)blockIdx.x * blockDim.x + threadIdx.x) >> 5);
  const int lane = threadIdx.x & 31;
  if (n >= NN) return;

  float4 yv = *reinterpret_cast<const float4*>(lin + (size_t)n * CC + lane * 4);
  float s  = yv.x + yv.y + yv.z + yv.w;
  float s2 = yv.x * yv.x + yv.y * yv.y + yv.z * yv.z + yv.w * yv.w;
  #pragma unroll
  for (int off = 16; off > 0; off >>= 1) {
    s  += __shfl_xor(s,  off, 32);
    s2 += __shfl_xor(s2, off, 32);
  }
  const float mu   = s * (1.0f / 128.0f);
  const float var  = s2 * (1.0f / 128.0f) - mu * mu;
  const float rstd = rsqrtf(var + EPSV);

  float4 hv = *reinterpret_cast<const float4*>(h0 + (size_t)n * CC + lane * 4);
  float4 gv = *reinterpret_cast<const float4*>(g1 + lane * 4);
  float4 bv = *reinterpret_cast<const float4*>(b1 + lane * 4);
  float4 r;
  r.x = hv.x + (yv.x - mu) * rstd * gv.x + bv.x;
  r.y = hv.y + (yv.y - mu) * rstd * gv.y + bv.y;
  r.z = hv.z + (yv.z - mu) * rstd * gv.z + bv.z;
  r.w = hv.w + (yv.w - mu) * rstd * gv.w + bv.w;
  *reinterpret_cast<float4*>(out + (size_t)n * CC + lane * 4) = r;
}

// =====================================================================
// host launcher
// =====================================================================
extern "C" void kernel_launch(void* const* d_in, const int* in_sizes, int n_in,
                              void* d_out, int out_size, void* d_ws, size_t ws_size,
                              hipStream_t stream) {
  const float* x     = (const float*)d_in[0];
  const int*   ei    = (const int*)  d_in[1];
  const float* eattr = (const float*)d_in[2];
  const float* Wq    = (const float*)d_in[3];
  const float* bq    = (const float*)d_in[4];
  const float* Wk    = (const float*)d_in[5];
  const float* bk    = (const float*)d_in[6];
  const float* Wv    = (const float*)d_in[7];
  const float* bv    = (const float*)d_in[8];
  const float* We    = (const float*)d_in[9];
  const float* Wskip = (const float*)d_in[10];
  const float* bskip = (const float*)d_in[11];
  const float* Wl    = (const float*)d_in[12];
  const float* bl    = (const float*)d_in[13];
  const float* g0    = (const float*)d_in[14];
  const float* b0    = (const float*)d_in[15];
  const float* g1    = (const float*)d_in[16];
  const float* b1    = (const float*)d_in[17];
  float* out = (float*)d_out;

  // workspace carve-up (floats)
  float* ws = (float*)d_ws;
  size_t off = 0;
  float* q     = ws + off; off += (size_t)NN * HC;     // 10.24M
  float* k     = ws + off; off += (size_t)NN * HC;
  float* v     = ws + off; off += (size_t)NN * HC;
  float* skip  = ws + off; off += (size_t)NN * CC;
  float* alpha = ws + off; off += (size_t)EE * HH;
  float* asum  = ws + off; off += (size_t)NN * HH;
  float* h0    = ws + off; off += (size_t)NN * CC;
  float* lin   = ws + off; off += (size_t)NN * CC;
  int*   amaxOrd = (int*)(ws + off); off += (size_t)NN * HH;
  float* accum = ws + off; off += (size_t)NN * HC;

  // 0) reset accumulators (every call -> deterministic)
  ts_init_kernel<<<2048, 256, 0, stream>>>(asum, amaxOrd, accum);

  // 1) projections (fp32 WMMA). N=20000 = 1250 row-tiles of 16.
  ts_wmma_gemm<<<dim3(1250, HC / 64), 128, 0, stream>>>(x, Wq, bq, q, HC);
  ts_wmma_gemm<<<dim3(1250, HC / 64), 128, 0, stream>>>(x, Wk, bk, k, HC);
  ts_wmma_gemm<<<dim3(1250, HC / 64), 128, 0, stream>>>(x, Wv, bv, v, HC);
  ts_wmma_gemm<<<dim3(1250, CC / 64), 128, 0, stream>>>(x, Wskip, bskip, skip, CC);

  // 2) edge attention scores + running max (one wave/edge; 8 waves/block)
  ts_edge_alpha<<<EE / 8, 256, 0, stream>>>(ei, eattr, q, k, We, alpha, amaxOrd);

  // 3) exp + weighted scatter (unnormalized) + denominator
  ts_edge_scatter<<<EE / 8, 256, 0, stream>>>(ei, eattr, v, We, alpha, amaxOrd,
                                              asum, accum);

  // 4) per-node normalize + head-mean + skip + LN0 + residual -> h0
  ts_node_combine<<<NN / 4, 128, 0, stream>>>(x, skip, accum, asum, g0, b0, h0);

  // 5) shell linear on h0 (fp32 WMMA)
  ts_wmma_gemm<<<dim3(1250, CC / 64), 128, 0, stream>>>(h0, Wl, bl, lin, CC);

  // 6) final LN + residual -> out
  ts_final_ln<<<NN / 4, 128, 0, stream>>>(h0, lin, g1, b1, out);
}